// MultipolarInteraction_10814727651503
// MI455X (gfx1250) — compile-verified
//
#include <hip/hip_runtime.h>
#include <math.h>

typedef float v2f __attribute__((ext_vector_type(2)));
typedef float v8f __attribute__((ext_vector_type(8)));

#define CUTOFF2_F      100.0f          // CUTOFF = 10.0
#define EWALD_ALPHA_F  0.3f
#define TWO_OVER_SQRTPI 1.12837916709551257390f

// ---------------------------------------------------------------------------
// Wave32 sum-reduction using the matrix core:
//   A (16x4): vgpr0 lanes0-15 -> K=0 (M=lane), lanes16-31 -> K=2 (M=lane-16);
//   vgpr1 = 0. B (4x16) = all ones.  D[m,n] = v[m] + v[m+16].
//   Per-lane sum of the 8 D vgprs gives half-wave totals (M=0..7 on lanes
//   0-15, M=8..15 on lanes 16-31); one xor-16 shuffle completes the sum.
// EXEC must be all-ones at the call site (callers guarantee wave-uniform
// control flow).
// ---------------------------------------------------------------------------
__device__ __forceinline__ float wave_reduce32(float v) {
    v2f a; a[0] = v;    a[1] = 0.0f;
    v2f b; b[0] = 1.0f; b[1] = 1.0f;
    v8f c = {};
    c = __builtin_amdgcn_wmma_f32_16x16x4_f32(false, a, false, b,
                                              (short)0, c, false, false);
    float s = c[0] + c[1] + c[2] + c[3] + c[4] + c[5] + c[6] + c[7];
    s += __shfl_xor(s, 16, 32);
    return s;   // every lane holds the full 32-lane sum
}

__device__ __forceinline__ float block_reduce(float v, float* sdata) {
    float ws = wave_reduce32(v);
    const int lane = threadIdx.x & 31;
    const int wid  = threadIdx.x >> 5;
    if (lane == 0) sdata[wid] = ws;
    __syncthreads();
    float tot = 0.0f;
    if (wid == 0) {
        const int nw = (blockDim.x + 31) >> 5;
        float x = (lane < nw) ? sdata[lane] : 0.0f;
        tot = wave_reduce32(x);
    }
    return tot;  // valid on wave 0
}

// ---------------------------------------------------------------------------
// Main kernel: one pair per lane, grid-stride. Atom data (6.4 MB) stays L2
// resident; only the pair list streams from HBM. ~98% of pairs fail the
// cutoff and skip the gather + tensor math entirely.
// ---------------------------------------------------------------------------
__global__ __launch_bounds__(256)
void pair_energy_kernel(const float* __restrict__ coords,
                        const float* __restrict__ box,
                        const int*   __restrict__ pairs,
                        const float* __restrict__ q,
                        const float* __restrict__ p,
                        const float* __restrict__ t,
                        float*       __restrict__ partials,
                        int npairs) {
    // ---- box and its inverse (uniform, hoisted out of the loop) ----
    const float b00 = box[0], b01 = box[1], b02 = box[2];
    const float b10 = box[3], b11 = box[4], b12 = box[5];
    const float b20 = box[6], b21 = box[7], b22 = box[8];
    const float det = b00 * (b11 * b22 - b12 * b21)
                    - b01 * (b10 * b22 - b12 * b20)
                    + b02 * (b10 * b21 - b11 * b20);
    const float id  = 1.0f / det;
    const float i00 =  (b11 * b22 - b12 * b21) * id;
    const float i01 = -(b01 * b22 - b02 * b21) * id;
    const float i02 =  (b01 * b12 - b02 * b11) * id;
    const float i10 = -(b10 * b22 - b12 * b20) * id;
    const float i11 =  (b00 * b22 - b02 * b20) * id;
    const float i12 = -(b00 * b12 - b02 * b10) * id;
    const float i20 =  (b10 * b21 - b11 * b20) * id;
    const float i21 = -(b00 * b21 - b01 * b20) * id;
    const float i22 =  (b00 * b11 - b01 * b10) * id;

    const int2* __restrict__ pr = (const int2*)pairs;

    float acc = 0.0f;
    const int stride = gridDim.x * blockDim.x;
    for (int pidx = blockIdx.x * blockDim.x + threadIdx.x;
         pidx < npairs; pidx += stride) {
        const int2 ij = pr[pidx];          // single b64 load
        const int ai = ij.x, aj = ij.y;

        float dx = coords[3 * aj + 0] - coords[3 * ai + 0];
        float dy = coords[3 * aj + 1] - coords[3 * ai + 1];
        float dz = coords[3 * aj + 2] - coords[3 * ai + 2];

        // minimum image: dr -= round(dr @ box_inv) @ box
        const float sx = dx * i00 + dy * i10 + dz * i20;
        const float sy = dx * i01 + dy * i11 + dz * i21;
        const float sz = dx * i02 + dy * i12 + dz * i22;
        const float nx = rintf(sx), ny = rintf(sy), nz = rintf(sz);
        dx -= nx * b00 + ny * b10 + nz * b20;
        dy -= nx * b01 + ny * b11 + nz * b21;
        dz -= nx * b02 + ny * b12 + nz * b22;

        const float r2 = dx * dx + dy * dy + dz * dz;
        if (r2 > CUTOFF2_F) continue;      // mask: ~98% of pairs exit here

        const float r    = sqrtf(r2);
        const float rinv = 1.0f / r;

        // ---- erfc damping factors ----
        const float u  = EWALD_ALPHA_F * r;
        const float u2 = u * u, u3 = u2 * u, u5 = u3 * u2, u7 = u5 * u2;
        const float e  = erfcf(u);
        const float g  = TWO_OVER_SQRTPI * expf(-u2);
        const float d0 = e;
        const float d1 = e + g * u;
        const float d2 = e + g * (3.0f * u + 2.0f * u3) * (1.0f / 3.0f);
        const float d3 = e + g * (15.0f * u + 10.0f * u3 + 4.0f * u5) * (1.0f / 15.0f);
        const float d4 = e + g * (8.0f * u7 + 28.0f * u5 + 70.0f * u3 + 105.0f * u)
                                 * (1.0f / 105.0f);

        const float ri2 = rinv * rinv;
        const float ri3 = ri2 * rinv, ri5 = ri3 * ri2;
        const float ri7 = ri5 * ri2,  ri9 = ri7 * ri2;
        const float D1 = rinv * d0, D3 = ri3 * d1, D5 = ri5 * d2;
        const float D7 = ri7 * d3,  D9 = ri9 * d4;

        // ---- interaction tensor components ----
        const float x = dx, y = dy, z = dz;
        const float x2 = x * x, y2 = y * y, z2 = z * z;
        const float xy = x * y, xz = x * z, yz = y * z;

        const float tx = -x * D3, ty = -y * D3, tz = -z * D3;
        const float txx = 3.0f * x2 * D5 - D3;
        const float txy = 3.0f * xy * D5;
        const float txz = 3.0f * xz * D5;
        const float tyy = 3.0f * y2 * D5 - D3;
        const float tyz = 3.0f * yz * D5;
        const float tzz = 3.0f * z2 * D5 - D3;
        const float txxx = -15.0f * x2 * x * D7 + 9.0f * x * D5;
        const float txxy = -15.0f * x2 * y * D7 + 3.0f * y * D5;
        const float txxz = -15.0f * x2 * z * D7 + 3.0f * z * D5;
        const float tyyy = -15.0f * y2 * y * D7 + 9.0f * y * D5;
        const float tyyx = -15.0f * y2 * x * D7 + 3.0f * x * D5;
        const float tyyz = -15.0f * y2 * z * D7 + 3.0f * z * D5;
        const float tzzz = -15.0f * z2 * z * D7 + 9.0f * z * D5;
        const float tzzx = -15.0f * z2 * x * D7 + 3.0f * x * D5;
        const float tzzy = -15.0f * z2 * y * D7 + 3.0f * y * D5;
        const float txyz = -15.0f * x * y * z * D7;
        const float txxxx = 105.0f * x2 * x2 * D9 - 90.0f * x2 * D7 + 9.0f * D5;
        const float txxxy = 105.0f * x2 * xy * D9 - 45.0f * xy * D7;
        const float txxxz = 105.0f * x2 * xz * D9 - 45.0f * xz * D7;
        const float txxyy = 105.0f * x2 * y2 * D9 - 15.0f * (x2 + y2) * D7 + 3.0f * D5;
        const float txxzz = 105.0f * x2 * z2 * D9 - 15.0f * (x2 + z2) * D7 + 3.0f * D5;
        const float txxyz = 105.0f * x2 * yz * D9 - 15.0f * yz * D7;
        const float tyyyy = 105.0f * y2 * y2 * D9 - 90.0f * y2 * D7 + 9.0f * D5;
        const float tyyyx = 105.0f * y2 * xy * D9 - 45.0f * xy * D7;
        const float tyyyz = 105.0f * y2 * yz * D9 - 45.0f * yz * D7;
        const float tyyzz = 105.0f * y2 * z2 * D9 - 15.0f * (y2 + z2) * D7 + 3.0f * D5;
        const float tyyxz = 105.0f * y2 * xz * D9 - 15.0f * xz * D7;
        const float tzzzz = 105.0f * z2 * z2 * D9 - 90.0f * z2 * D7 + 9.0f * D5;
        const float tzzzx = 105.0f * z2 * xz * D9 - 45.0f * xz * D7;
        const float tzzzy = 105.0f * z2 * yz * D9 - 45.0f * yz * D7;
        const float tzzxy = 105.0f * z2 * xy * D9 - 15.0f * xy * D7;

        // ---- packed multipoles (gathered only for surviving pairs) ----
        const float mi0 = q[ai];
        const float* pv = p + 3 * ai;
        const float mi1 = pv[0], mi2 = pv[1], mi3 = pv[2];
        const float* tv = t + 9 * ai;
        const float mi4 = tv[0] * (1.0f / 3.0f);
        const float mi5 = (tv[1] + tv[3]) * (1.0f / 3.0f);
        const float mi6 = (tv[2] + tv[6]) * (1.0f / 3.0f);
        const float mi7 = tv[4] * (1.0f / 3.0f);
        const float mi8 = (tv[5] + tv[7]) * (1.0f / 3.0f);
        const float mi9 = tv[8] * (1.0f / 3.0f);

        const float mj0 = q[aj];
        const float* pw = p + 3 * aj;
        const float mj1 = pw[0], mj2 = pw[1], mj3 = pw[2];
        const float* tw = t + 9 * aj;
        const float mj4 = tw[0] * (1.0f / 3.0f);
        const float mj5 = (tw[1] + tw[3]) * (1.0f / 3.0f);
        const float mj6 = (tw[2] + tw[6]) * (1.0f / 3.0f);
        const float mj7 = tw[4] * (1.0f / 3.0f);
        const float mj8 = (tw[5] + tw[7]) * (1.0f / 3.0f);
        const float mj9 = tw[8] * (1.0f / 3.0f);

        // ---- v = T * m_i (rows of the 10x10, never materialized) ----
        const float v0 = D1*mi0   - tx*mi1    - ty*mi2    - tz*mi3
                       + txx*mi4  + txy*mi5   + txz*mi6   + tyy*mi7
                       + tyz*mi8  + tzz*mi9;
        const float v1 = tx*mi0   - txx*mi1   - txy*mi2   - txz*mi3
                       + txxx*mi4 + txxy*mi5  + txxz*mi6  + tyyx*mi7
                       + txyz*mi8 + tzzx*mi9;
        const float v2 = ty*mi0   - txy*mi1   - tyy*mi2   - tyz*mi3
                       + txxy*mi4 + tyyx*mi5  + txyz*mi6  + tyyy*mi7
                       + tyyz*mi8 + tzzy*mi9;
        const float v3 = tz*mi0   - txz*mi1   - tyz*mi2   - tzz*mi3
                       + txxz*mi4 + txyz*mi5  + tzzx*mi6  + tyyz*mi7
                       + tzzy*mi8 + tzzz*mi9;
        const float v4 = txx*mi0  - txxx*mi1  - txxy*mi2  - txxz*mi3
                       + txxxx*mi4 + txxxy*mi5 + txxxz*mi6 + txxyy*mi7
                       + txxyz*mi8 + txxzz*mi9;
        const float v5 = txy*mi0  - txxy*mi1  - tyyx*mi2  - txyz*mi3
                       + txxxy*mi4 + txxyy*mi5 + txxyz*mi6 + tyyyx*mi7
                       + tyyxz*mi8 + tzzxy*mi9;
        const float v6 = txz*mi0  - txxz*mi1  - txyz*mi2  - tzzx*mi3
                       + txxxz*mi4 + txxyz*mi5 + txxzz*mi6 + tyyxz*mi7
                       + tzzxy*mi8 + tzzzx*mi9;
        const float v7 = tyy*mi0  - tyyx*mi1  - tyyy*mi2  - tyyz*mi3
                       + txxyy*mi4 + tyyyx*mi5 + tyyxz*mi6 + tyyyy*mi7
                       + tyyyz*mi8 + tyyzz*mi9;
        const float v8 = tyz*mi0  - txyz*mi1  - tyyz*mi2  - tzzy*mi3
                       + txxyz*mi4 + tyyxz*mi5 + tzzxy*mi6 + tyyyz*mi7
                       + tyyzz*mi8 + tzzzy*mi9;
        const float v9 = tzz*mi0  - tzzx*mi1  - tzzy*mi2  - tzzz*mi3
                       + txxzz*mi4 + tzzxy*mi5 + tzzzx*mi6 + tyyzz*mi7
                       + tzzzy*mi8 + tzzzz*mi9;

        acc += mj0*v0 + mj1*v1 + mj2*v2 + mj3*v3 + mj4*v4
             + mj5*v5 + mj6*v6 + mj7*v7 + mj8*v8 + mj9*v9;
    }

    __shared__ float sdata[32];
    const float tot = block_reduce(acc, sdata);
    if (threadIdx.x == 0) partials[blockIdx.x] = tot;
}

// ---------------------------------------------------------------------------
// Deterministic final reduction: single block folds per-block partials.
// ---------------------------------------------------------------------------
__global__ __launch_bounds__(256)
void final_reduce_kernel(const float* __restrict__ partials, int n,
                         float* __restrict__ out) {
    float acc = 0.0f;
    for (int i = threadIdx.x; i < n; i += blockDim.x) acc += partials[i];
    __shared__ float sdata[32];
    const float tot = block_reduce(acc, sdata);
    if (threadIdx.x == 0) out[0] = tot;   // PREFACTOR == 1.0
}

extern "C" void kernel_launch(void* const* d_in, const int* in_sizes, int n_in,
                              void* d_out, int out_size, void* d_ws, size_t ws_size,
                              hipStream_t stream) {
    const float* coords = (const float*)d_in[0];
    const float* box    = (const float*)d_in[1];
    const int*   pairs  = (const int*)d_in[2];
    const float* q      = (const float*)d_in[3];
    const float* p      = (const float*)d_in[4];
    const float* t      = (const float*)d_in[5];
    float*       out    = (float*)d_out;

    const int npairs = in_sizes[2] / 2;

    const int block = 256;
    int nblocks = 2048;                                    // persistent grid
    const int need = (npairs + block - 1) / block;
    if (nblocks > need) nblocks = need;
    const int maxblk = (int)(ws_size / sizeof(float));
    if (nblocks > maxblk) nblocks = maxblk;
    if (nblocks < 1) nblocks = 1;

    float* partials = (float*)d_ws;
    pair_energy_kernel<<<nblocks, block, 0, stream>>>(coords, box, pairs,
                                                      q, p, t, partials, npairs);
    final_reduce_kernel<<<1, 256, 0, stream>>>(partials, nblocks, out);
}